// Loss4RobustPixelReconstructionByGridSamplingOffset_85203561218677
// MI455X (gfx1250) — compile-verified
//
#include <hip/hip_runtime.h>

typedef __attribute__((ext_vector_type(2))) float v2f;
typedef __attribute__((ext_vector_type(8))) float v8f;

#define BDIM 256
#define NBLK 2048
#define B_   16
#define C_   3
#define H_   768
#define W_   768
#define HW_  (H_ * W_)                 // 589824
#define NPIX (B_ * HW_)                // 9437184  (== NBLK*BDIM*18 exactly)
#define NELEM (B_ * C_ * HW_)          // 28311552
#define EPS2 1.0e-6f                   // EPS*EPS, EPS = 0.001
#define MASK_THRESH 0.9999f

// Full-wave (32 lane) f32 sum via V_WMMA_F32_16X16X4_F32.
// A: a0 = p (lanes 0-15 -> A[m,0], lanes 16-31 -> A[m,2]), a1 = 0.
// B: all ones (layout-agnostic).  D[m,n] = p_m + p_{m+16} for every n.
// Per-lane sum of the 8 accumulator VGPRs covers rows 0-7 (lanes<16) or
// 8-15 (lanes>=16); one shfl_xor(16) completes the reduction.
// Caller must guarantee EXEC is all ones (no divergence at call site).
__device__ __forceinline__ float wave_sum_wmma(float p) {
    v2f a; a[0] = p;    a[1] = 0.0f;
    v2f b; b[0] = 1.0f; b[1] = 1.0f;
    v8f c = {};
    c = __builtin_amdgcn_wmma_f32_16x16x4_f32(
        /*neg_a=*/false, a, /*neg_b=*/false, b,
        /*c_mod=*/(short)0, c, /*reuse_a=*/false, /*reuse_b=*/false);
    float s = c[0] + c[1] + c[2] + c[3] + c[4] + c[5] + c[6] + c[7];
    s += __shfl_xor(s, 16, 32);
    return s;
}

// Block-wide sum (valid in thread 0). BDIM = 256 -> 8 wave32 waves.
__device__ __forceinline__ float block_sum(float p, float* lds) {
    float w = wave_sum_wmma(p);
    const int lane = threadIdx.x & 31;
    const int wid  = threadIdx.x >> 5;
    if (lane == 0) lds[wid] = w;
    __syncthreads();
    float r = 0.0f;
    if (threadIdx.x == 0) {
#pragma unroll
        for (int i = 0; i < BDIM / 32; ++i) r += lds[i];
    }
    return r;
}

__global__ void __launch_bounds__(BDIM)
warp_charb_kernel(const float* __restrict__ img1,
                  const float* __restrict__ img2,
                  const float* __restrict__ uv,
                  float* __restrict__ partials) {
    __shared__ float lds[BDIM / 32];
    const int tid    = blockIdx.x * BDIM + threadIdx.x;
    const int stride = NBLK * BDIM;

    float acc = 0.0f;
    // NPIX is an exact multiple of stride (18 iterations): zero divergence,
    // EXEC stays all-ones through the trailing WMMA reduction.
    for (int i = tid; i < NPIX; i += stride) {
        const int b   = i / HW_;
        const int rem = i - b * HW_;
        const int y   = rem / W_;
        const int x   = rem - y * W_;

        // Streamed-once data: non-temporal so the reused image1 gather
        // source stays resident in the 192MB L2.
        const float* uvb = uv + (size_t)b * 2 * HW_;
        const float fx = __builtin_nontemporal_load(uvb + rem);
        const float fy = __builtin_nontemporal_load(uvb + HW_ + rem);

        // Prefetch next grid-stride iteration's streaming reads (the large
        // stride defeats sequential prefetch). Uniform predicate per wave.
        const int ni = i + stride;
        if (ni < NPIX) {
            const int nb   = ni / HW_;
            const int nrem = ni - nb * HW_;
            __builtin_prefetch(uv   + (size_t)nb * 2 * HW_ + nrem, 0, 0);
            __builtin_prefetch(img2 + (size_t)nb * 3 * HW_ + nrem, 0, 0);
        }

        const float px = (float)x + fx;
        const float py = (float)y + fy;
        const float x0f = floorf(px), y0f = floorf(py);
        const float wx1 = px - x0f,   wy1 = py - y0f;
        const float wx0 = 1.0f - wx1, wy0 = 1.0f - wy1;

        const int ix0 = (int)x0f, iy0 = (int)y0f;
        const int ix1 = ix0 + 1,  iy1 = iy0 + 1;

        const float bx0 = (ix0 >= 0 && ix0 < W_) ? 1.0f : 0.0f;
        const float bx1 = (ix1 >= 0 && ix1 < W_) ? 1.0f : 0.0f;
        const float by0 = (iy0 >= 0 && iy0 < H_) ? 1.0f : 0.0f;
        const float by1 = (iy1 >= 0 && iy1 < H_) ? 1.0f : 0.0f;

        const float w00 = wx0 * wy0 * bx0 * by0;
        const float w10 = wx1 * wy0 * bx1 * by0;
        const float w01 = wx0 * wy1 * bx0 * by1;
        const float w11 = wx1 * wy1 * bx1 * by1;

        const float msum = w00 + w10 + w01 + w11;
        const float mask = (msum < MASK_THRESH) ? 0.0f : 1.0f;

        const int cx0 = ix0 < 0 ? 0 : (ix0 > W_ - 1 ? W_ - 1 : ix0);
        const int cx1 = ix1 < 0 ? 0 : (ix1 > W_ - 1 ? W_ - 1 : ix1);
        const int cy0 = iy0 < 0 ? 0 : (iy0 > H_ - 1 ? H_ - 1 : iy0);
        const int cy1 = iy1 < 0 ? 0 : (iy1 > H_ - 1 ? H_ - 1 : iy1);

        const int o00 = cy0 * W_ + cx0;
        const int o10 = cy0 * W_ + cx1;
        const int o01 = cy1 * W_ + cx0;
        const int o11 = cy1 * W_ + cx1;

        const float* i1b = img1 + (size_t)b * 3 * HW_;
        const float* i2b = img2 + (size_t)b * 3 * HW_ + rem;

#pragma unroll
        for (int ch = 0; ch < 3; ++ch) {
            const float* pc = i1b + ch * HW_;
            const float v = w00 * pc[o00] + w10 * pc[o10]
                          + w01 * pc[o01] + w11 * pc[o11];
            const float t = __builtin_nontemporal_load(i2b + ch * HW_);
            const float d = v * mask - t;
            acc += sqrtf(fmaf(d, d, EPS2));
        }
    }

    const float bs = block_sum(acc, lds);
    if (threadIdx.x == 0) partials[blockIdx.x] = bs;
}

__global__ void __launch_bounds__(BDIM)
reduce_kernel(const float* __restrict__ partials, float* __restrict__ out) {
    __shared__ float lds[BDIM / 32];
    float acc = 0.0f;
    // NBLK / BDIM = 8 exact iterations: no divergence before the WMMA.
    for (int i = threadIdx.x; i < NBLK; i += BDIM)
        acc += partials[i];
    const float s = block_sum(acc, lds);
    if (threadIdx.x == 0)
        out[0] = s * (1.0f / (float)NELEM);
}

extern "C" void kernel_launch(void* const* d_in, const int* in_sizes, int n_in,
                              void* d_out, int out_size, void* d_ws, size_t ws_size,
                              hipStream_t stream) {
    const float* img1 = (const float*)d_in[0];
    const float* img2 = (const float*)d_in[1];
    const float* uv   = (const float*)d_in[2];
    float* partials   = (float*)d_ws;   // NBLK floats (8 KB)

    warp_charb_kernel<<<NBLK, BDIM, 0, stream>>>(img1, img2, uv, partials);
    reduce_kernel<<<1, BDIM, 0, stream>>>(partials, (float*)d_out);
}